// TMessagePassing_12128987644196
// MI455X (gfx1250) — compile-verified
//
#include <hip/hip_runtime.h>
#include <stdint.h>

// TMessagePassing aggregate: out[b,:] = sum_c w[b,c] * (F[i0(b,c),:] * F[i1(b,c),:]),
// falling back to F[target[b],:] when the node has no incident edges.
//
// MI455X reasoning: ~0.1 GFLOP vs ~250MB of row gathers -> deeply memory bound;
// features (25.6MB) + metadata + out are all L2-resident (192MB L2), so this is
// an L2-latency-bound gather kernel. No matmul structure -> WMMA does not apply.
//
// CDNA5 paths used:
//  - GLOBAL_LOAD_ASYNC_TO_LDS_B128 (ASYNCcnt) in GVS mode (SGPR base + 32-bit
//    VGPR offset) keeps a depth-4 ring of combo row-pairs in flight per wave
//    with no VGPR cost; consumed via ds_load_b128.
//  - readfirstlane forces all per-combo metadata onto the scalar path
//    (s_load_b32/b64, KMcnt) since b is wave-uniform under wave32.
//  - global_prefetch_b8 streams metadata ahead.

#define WAVE_SZ 32
#define WAVES_PER_BLOCK 8
#define BLOCK_THREADS (WAVE_SZ * WAVES_PER_BLOCK)
#define DEPTH 4                       // combo slots in flight per wave
#define ROW_BYTES 512                 // D=128 floats
#define SLOT_BYTES (2 * ROW_BYTES)    // two rows per combo

// Issue both row gathers of combo (i0,i1) into the wave's LDS ring slot.
// GVS mode: 64-bit feature base in SGPRs, 32-bit per-lane byte offset in VGPR.
// s_wait_dscnt 0 first: the slot being overwritten was ds_read last iteration.
__device__ __forceinline__ void issue_combo_async(uint64_t fbase, int i0, int i1,
                                                  uint32_t slot_lds, int lane)
{
    const uint32_t laneB = (uint32_t)(lane * 16);
    const uint32_t o0 = ((uint32_t)i0 << 9) + laneB;   // idx * 512B row + lane chunk
    const uint32_t o1 = ((uint32_t)i1 << 9) + laneB;
    const uint32_t l0 = slot_lds + laneB;
    const uint32_t l1 = l0 + ROW_BYTES;
    asm volatile("s_wait_dscnt 0x0\n\t"
                 "global_load_async_to_lds_b128 %0, %2, %4\n\t"
                 "global_load_async_to_lds_b128 %1, %3, %4"
                 :: "v"(l0), "v"(l1), "v"(o0), "v"(o1), "s"(fbase)
                 : "memory");
}

// ---------------- D == 128 fast path: async-to-LDS pipelined gather ----------------
__global__ __launch_bounds__(BLOCK_THREADS)
void tmp_aggregate_async_kernel(const float* __restrict__ features,
                                const int*   __restrict__ target_nodes,
                                const int*   __restrict__ comb_idx,   // [B, cmax, 2]
                                const float* __restrict__ comb_w,     // [B, cmax]
                                const float* __restrict__ has_edge,   // [B]
                                float*       __restrict__ out,        // [B, 128]
                                int B, int cmax)
{
    constexpr int D = 128;
    __shared__ float ring[WAVES_PER_BLOCK][DEPTH][2][D];   // 32 KB

    const int wave = threadIdx.x >> 5;
    const int lane = threadIdx.x & (WAVE_SZ - 1);
    const int b    = blockIdx.x * WAVES_PER_BLOCK + wave;
    if (b >= B) return;

    // b is wave-uniform under wave32 but LLVM's divergence analysis can't see
    // it; readfirstlane moves all metadata addressing to the scalar pipe.
    const int bu = __builtin_amdgcn_readfirstlane(b);

    const long long base = (long long)bu * cmax;
    const float* __restrict__ wptr = comb_w + base;
    const int2*  __restrict__ iptr = (const int2*)comb_idx + base;
    const uint64_t fbase = (uint64_t)(uintptr_t)features;

    // Flat LDS address = aperture | offset; hardware truncates to addr[31:0].
    const uint32_t wave_lds = (uint32_t)(uintptr_t)(&ring[wave][0][0][0]);

    // Prologue: put DEPTH-1 combos in flight (padding combos have idx 0 -> safe).
    #pragma unroll
    for (int p = 0; p < DEPTH - 1; ++p)
        if (p < cmax) {
            const int2 id = iptr[p];                    // scalar s_load_b64
            issue_combo_async(fbase, id.x, id.y,
                              wave_lds + (uint32_t)p * SLOT_BYTES, lane);
        }

    float4 acc = make_float4(0.f, 0.f, 0.f, 0.f);

    for (int c = 0; c < cmax; ++c) {
        const float w = wptr[c];                        // scalar s_load_b32

        // Stream-prefetch metadata ahead (gfx1250 global_prefetch_b8).
        if (((c & 15) == 0) && (c + 32 < cmax)) {
            __builtin_prefetch(wptr + c + 32, 0, 0);
            __builtin_prefetch(iptr + c + 32, 0, 0);
        }

        if (c + DEPTH - 1 < cmax) {
            const int2 idn = iptr[c + DEPTH - 1];       // scalar s_load_b64
            issue_combo_async(fbase, idn.x, idn.y,
                              wave_lds + (uint32_t)((c + DEPTH - 1) & (DEPTH - 1)) * SLOT_BYTES,
                              lane);
            // 2 ops/combo * (DEPTH-1) newer combos may remain in flight
            // -> combo c's loads (in-order completion) are done.
            asm volatile("s_wait_asynccnt 0x6" ::: "memory");
        } else {
            asm volatile("s_wait_asynccnt 0x0" ::: "memory");  // drain tail
        }

        if (w == 0.0f) break;               // trailing zero weights == padding

        const int s = c & (DEPTH - 1);
        const float4 f0 = ((const float4*)&ring[wave][s][0][0])[lane];  // ds_load_b128
        const float4 f1 = ((const float4*)&ring[wave][s][1][0])[lane];

        acc.x = fmaf(w, f0.x * f1.x, acc.x);
        acc.y = fmaf(w, f0.y * f1.y, acc.y);
        acc.z = fmaf(w, f0.z * f1.z, acc.z);
        acc.w = fmaf(w, f0.w * f1.w, acc.w);
    }

    float4 res = acc;
    if (has_edge[bu] <= 0.0f) {             // isolated node: return own feature row
        const int t = target_nodes[bu];     // scalar
        res = *(const float4*)(features + (long long)t * D + lane * 4);
    }
    *(float4*)(out + (long long)b * D + lane * 4) = res;
}

// ---------------- generic fallback (any D % 4 == 0): direct VGPR gather ----------------
__global__ __launch_bounds__(BLOCK_THREADS)
void tmp_aggregate_kernel(const float* __restrict__ features,
                          const int*   __restrict__ target_nodes,
                          const int*   __restrict__ comb_idx,
                          const float* __restrict__ comb_w,
                          const float* __restrict__ has_edge,
                          float*       __restrict__ out,
                          int B, int D, int cmax)
{
    const int wave = threadIdx.x >> 5;
    const int lane = threadIdx.x & (WAVE_SZ - 1);
    const int b    = blockIdx.x * WAVES_PER_BLOCK + wave;
    if (b >= B) return;

    const int dOff = blockIdx.y * (WAVE_SZ * 4) + lane * 4;
    if (dOff >= D) return;

    const int bu = __builtin_amdgcn_readfirstlane(b);
    const long long base = (long long)bu * cmax;
    const float* __restrict__ wptr = comb_w + base;
    const int2*  __restrict__ iptr = (const int2*)comb_idx + base;

    float4 acc = make_float4(0.f, 0.f, 0.f, 0.f);
    float w = wptr[0];
    if (w != 0.0f) {
        int2 id = iptr[0];
        float4 f0 = *(const float4*)(features + (long long)id.x * D + dOff);
        float4 f1 = *(const float4*)(features + (long long)id.y * D + dOff);
        int c = 0;
        for (;;) {
            float  wn  = 0.0f;
            float4 nf0 = make_float4(0.f, 0.f, 0.f, 0.f);
            float4 nf1 = nf0;
            const int cn = c + 1;
            if (cn < cmax) {
                wn = wptr[cn];
                if (wn != 0.0f) {
                    int2 idn = iptr[cn];
                    nf0 = *(const float4*)(features + (long long)idn.x * D + dOff);
                    nf1 = *(const float4*)(features + (long long)idn.y * D + dOff);
                }
            }
            acc.x = fmaf(w, f0.x * f1.x, acc.x);
            acc.y = fmaf(w, f0.y * f1.y, acc.y);
            acc.z = fmaf(w, f0.z * f1.z, acc.z);
            acc.w = fmaf(w, f0.w * f1.w, acc.w);
            if (wn == 0.0f) break;
            w = wn; f0 = nf0; f1 = nf1; c = cn;
        }
    }

    float4 res = acc;
    if (has_edge[bu] <= 0.0f) {
        const int t = target_nodes[bu];
        res = *(const float4*)(features + (long long)t * D + dOff);
    }
    *(float4*)(out + (long long)b * D + dOff) = res;
}

extern "C" void kernel_launch(void* const* d_in, const int* in_sizes, int n_in,
                              void* d_out, int out_size, void* d_ws, size_t ws_size,
                              hipStream_t stream) {
    (void)n_in; (void)d_ws; (void)ws_size;

    const float* features     = (const float*)d_in[0];
    const int*   target_nodes = (const int*)  d_in[1];
    const int*   comb_idx     = (const int*)  d_in[2];
    const float* comb_w       = (const float*)d_in[3];
    const float* has_edge     = (const float*)d_in[4];
    float*       out          = (float*)d_out;

    const int B    = in_sizes[1];
    const int cmax = in_sizes[3] / B;
    const int D    = out_size / B;

    if (D == 128) {
        dim3 block(BLOCK_THREADS);
        dim3 grid((B + WAVES_PER_BLOCK - 1) / WAVES_PER_BLOCK);
        hipLaunchKernelGGL(tmp_aggregate_async_kernel, grid, block, 0, stream,
                           features, target_nodes, comb_idx, comb_w, has_edge, out,
                           B, cmax);
    } else {
        dim3 block(BLOCK_THREADS);
        dim3 grid((B + WAVES_PER_BLOCK - 1) / WAVES_PER_BLOCK,
                  (D + (WAVE_SZ * 4) - 1) / (WAVE_SZ * 4));
        hipLaunchKernelGGL(tmp_aggregate_kernel, grid, block, 0, stream,
                           features, target_nodes, comb_idx, comb_w, has_edge, out,
                           B, D, cmax);
    }
}